// GAT_gen_69286412419514
// MI455X (gfx1250) — compile-verified
//
#include <hip/hip_runtime.h>
#include <math.h>

typedef float v2f __attribute__((ext_vector_type(2)));
typedef float v8f __attribute__((ext_vector_type(8)));

#define NEG_SLOPE 0.2f
#define ENC_NEG_INF 0x007FFFFFu

// Order-preserving map float -> uint32 so unsigned atomicMax == float max.
__device__ __forceinline__ unsigned fenc(float f) {
  unsigned u = __float_as_uint(f);
  return (u & 0x80000000u) ? ~u : (u | 0x80000000u);
}
__device__ __forceinline__ float fdec(unsigned k) {
  unsigned u = (k & 0x80000000u) ? (k & 0x7FFFFFFFu) : ~k;
  return __uint_as_float(u);
}

// ---------------------------------------------------------------------------
// H = X @ W   (X:[N,128], W:[128,128], H:[N,128])  via V_WMMA_F32_16X16X4_F32
// Block = 256 threads = 8 waves; wave w computes the 16x16 tile
// rows [blockIdx.x*16, +16) x cols [w*16, +16); K=128 -> 32 chained WMMAs.
// ISA lane layouts (CDNA5 ISA):
//   A 16x4 : lanes0-15 M=lane,K={0,1} in v0,v1; lanes16-31 K={2,3}
//   B 4x16 : vgpr j, lanes0-15 K=j, lanes16-31 K=2+j; N=lane&15
//   C 16x16: vgpr v,  lanes0-15 M=v, lanes16-31 M=8+v; N=lane&15
// ---------------------------------------------------------------------------
__global__ void gat_gemm128_wmma(const float* __restrict__ X,
                                 const float* __restrict__ W,
                                 float* __restrict__ H, int N) {
  const int wave = threadIdx.x >> 5;
  const int lane = threadIdx.x & 31;
  const int half = lane >> 4;        // 0: lanes 0-15, 1: lanes 16-31
  const int lm   = lane & 15;
  const int row0 = blockIdx.x << 4;
  const int col  = (wave << 4) + lm;

  int arow = row0 + lm;
  if (arow >= N) arow = N - 1;       // N%16==0 here; clamp keeps EXEC all-1s
  const float* xr = X + (size_t)arow * 128;

  v8f c = {};
#pragma unroll
  for (int kk = 0; kk < 32; ++kk) {
    const int kb = (kk << 2) + (half << 1);   // this half-wave's K base
    v2f a, b;
    a.x = xr[kb];
    a.y = xr[kb + 1];
    b.x = W[(size_t)kb * 128 + col];
    b.y = W[(size_t)(kb + 1) * 128 + col];
    c = __builtin_amdgcn_wmma_f32_16x16x4_f32(false, a, false, b,
                                              (short)0, c, false, false);
  }

  const int orow0 = row0 + (half << 3);
#pragma unroll
  for (int v = 0; v < 8; ++v) {
    int r = orow0 + v;
    if (r < N) H[(size_t)r * 128 + col] = c[v];
  }
}

// ---------------------------------------------------------------------------
// Per-node attention logits: als[n,h] = sum_c H[n,h,c]*a_src[h,c] (same a_dst).
// One wave32 per node; lanes 0-15 own head 0, lanes 16-31 head 1.
// ---------------------------------------------------------------------------
__global__ void gat_attn_logits(const float* __restrict__ H,
                                const float* __restrict__ a_src,
                                const float* __restrict__ a_dst,
                                float* __restrict__ als,
                                float* __restrict__ ald, int N) {
  const int wid  = blockIdx.x * (blockDim.x >> 5) + (threadIdx.x >> 5);
  const int lane = threadIdx.x & 31;
  if (wid >= N) return;                         // wave-uniform
  const float* hr = H + (size_t)wid * 128 + lane * 4;
  const float* as = a_src + lane * 4;
  const float* ad = a_dst + lane * 4;
  float s = 0.f, d = 0.f;
#pragma unroll
  for (int j = 0; j < 4; ++j) {
    float hv = hr[j];
    s += hv * as[j];
    d += hv * ad[j];
  }
#pragma unroll
  for (int m = 8; m >= 1; m >>= 1) {            // reduce within each 16-group
    s += __shfl_xor(s, m, 32);
    d += __shfl_xor(d, m, 32);
  }
  if ((lane & 15) == 0) {
    const int h = lane >> 4;
    als[wid * 2 + h] = s;
    ald[wid * 2 + h] = d;
  }
}

// Zero accumulators / denominators, seed segment-max with encoded -inf.
__global__ void gat_init(float* __restrict__ agg, unsigned* __restrict__ emax,
                         float* __restrict__ denom, int N) {
  const int tid = blockIdx.x * blockDim.x + threadIdx.x;
  if (tid < N * 128) agg[tid] = 0.f;
  if (tid < N * 2) { emax[tid] = ENC_NEG_INF; denom[tid] = 0.f; }
}

__device__ __forceinline__ void edge_sd(int e, int E, const int* __restrict__ es,
                                        const int* __restrict__ ed, int& s, int& d) {
  if (e < E) { s = es[e]; d = ed[e]; }
  else       { s = e - E; d = e - E; }          // appended self-loops
}

// Pass 1: segment max of leaky_relu(als[src]+ald[dst]) over dst.
__global__ void gat_edge_max(const int* __restrict__ es, const int* __restrict__ ed,
                             const float* __restrict__ als, const float* __restrict__ ald,
                             unsigned* __restrict__ emax, int E, int Etot) {
  const int e = blockIdx.x * blockDim.x + threadIdx.x;
  if (e >= Etot) return;
  int s, d; edge_sd(e, E, es, ed, s, d);
#pragma unroll
  for (int h = 0; h < 2; ++h) {
    float v = als[s * 2 + h] + ald[d * 2 + h];
    v = (v >= 0.f) ? v : NEG_SLOPE * v;
    atomicMax(&emax[d * 2 + h], fenc(v));
  }
}

// Pass 2: softmax denominator per dst.
__global__ void gat_edge_sum(const int* __restrict__ es, const int* __restrict__ ed,
                             const float* __restrict__ als, const float* __restrict__ ald,
                             const unsigned* __restrict__ emax,
                             float* __restrict__ denom, int E, int Etot) {
  const int e = blockIdx.x * blockDim.x + threadIdx.x;
  if (e >= Etot) return;
  int s, d; edge_sd(e, E, es, ed, s, d);
#pragma unroll
  for (int h = 0; h < 2; ++h) {
    float v = als[s * 2 + h] + ald[d * 2 + h];
    v = (v >= 0.f) ? v : NEG_SLOPE * v;
    atomicAdd(&denom[d * 2 + h], expf(v - fdec(emax[d * 2 + h])));
  }
}

// Pass 3: agg[dst] += alpha * H[src]. One wave32 per edge (4 floats/lane),
// grid-strided with a prefetch of the next edge's source row.
__global__ void gat_edge_agg(const int* __restrict__ es, const int* __restrict__ ed,
                             const float* __restrict__ als, const float* __restrict__ ald,
                             const unsigned* __restrict__ emax,
                             const float* __restrict__ denom,
                             const float* __restrict__ H, float* __restrict__ agg,
                             int E, int Etot, int nWaves) {
  const int wid  = blockIdx.x * (blockDim.x >> 5) + (threadIdx.x >> 5);
  const int lane = threadIdx.x & 31;
  for (int e = wid; e < Etot; e += nWaves) {
    int s, d; edge_sd(e, E, es, ed, s, d);
    const int en = e + nWaves;                  // prefetch next gather row
    if (en < Etot) {
      const int sn = (en < E) ? es[en] : (en - E);
      __builtin_prefetch(&H[(size_t)sn * 128 + lane * 4], 0, 0);
    }
    float alpha[2];
#pragma unroll
    for (int h = 0; h < 2; ++h) {
      float v = als[s * 2 + h] + ald[d * 2 + h];
      v = (v >= 0.f) ? v : NEG_SLOPE * v;
      alpha[h] = expf(v - fdec(emax[d * 2 + h])) / denom[d * 2 + h];
    }
    const float* hs = H + (size_t)s * 128;
    float* ag = agg + (size_t)d * 128;
#pragma unroll
    for (int k = 0; k < 4; ++k) {
      const int idx = lane + (k << 5);
      atomicAdd(&ag[idx], hs[idx] * ((idx < 64) ? alpha[0] : alpha[1]));
    }
  }
}

// Layer-1 epilogue: in-place agg = ELU(agg + b1), b1:[128].
__global__ void gat_bias_elu(float* __restrict__ agg, const float* __restrict__ b, int N) {
  const int tid = blockIdx.x * blockDim.x + threadIdx.x;
  if (tid >= N * 128) return;
  const float v = agg[tid] + b[tid & 127];
  agg[tid] = (v > 0.f) ? v : (expf(v) - 1.f);
}

// Layer-2 epilogue: out[n,c] = mean over heads + b2, b2:[64].
__global__ void gat_head_mean_bias(const float* __restrict__ agg,
                                   const float* __restrict__ b,
                                   float* __restrict__ out, int N) {
  const int tid = blockIdx.x * blockDim.x + threadIdx.x;
  if (tid >= N * 64) return;
  const int n = tid >> 6, c = tid & 63;
  out[tid] = 0.5f * (agg[(size_t)n * 128 + c] + agg[(size_t)n * 128 + 64 + c]) + b[c];
}

extern "C" void kernel_launch(void* const* d_in, const int* in_sizes, int n_in,
                              void* d_out, int out_size, void* d_ws, size_t ws_size,
                              hipStream_t stream) {
  const float* x   = (const float*)d_in[0];
  const int*   ei  = (const int*)d_in[1];     // [2,E]; JAX x64-off -> int32
  const float* W1  = (const float*)d_in[2];
  const float* as1 = (const float*)d_in[3];
  const float* ad1 = (const float*)d_in[4];
  const float* b1  = (const float*)d_in[5];
  const float* W2  = (const float*)d_in[6];
  const float* as2 = (const float*)d_in[7];
  const float* ad2 = (const float*)d_in[8];
  const float* b2  = (const float*)d_in[9];

  const int N = in_sizes[0] / 128;
  const int E = in_sizes[1] / 2;
  const int Etot = E + N;
  const int* esrc = ei;
  const int* edst = ei + E;

  // Workspace layout (~106 MB): h | agg | als | ald | emax | denom
  float*    h     = (float*)d_ws;
  float*    agg   = h + (size_t)N * 128;
  float*    als   = agg + (size_t)N * 128;
  float*    ald   = als + (size_t)N * 2;
  unsigned* emax  = (unsigned*)(ald + (size_t)N * 2);
  float*    denom = (float*)(emax + (size_t)N * 2);

  const int B = 256;                       // 8 wave32s per block
  const int gGemm  = (N + 15) / 16;
  const int gNodeW = (N + 7) / 8;
  const int gInit  = (N * 128 + B - 1) / B;
  const int gEdge  = (Etot + B - 1) / B;
  const int aggBlocks = 8192;
  const int nWaves = aggBlocks * (B / 32);
  const int gFin1 = (N * 128 + B - 1) / B;
  const int gFin2 = (N * 64 + B - 1) / B;

  // ---- Layer 1 (concat heads) ----
  gat_gemm128_wmma<<<gGemm, B, 0, stream>>>(x, W1, h, N);
  gat_attn_logits<<<gNodeW, B, 0, stream>>>(h, as1, ad1, als, ald, N);
  gat_init<<<gInit, B, 0, stream>>>(agg, emax, denom, N);
  gat_edge_max<<<gEdge, B, 0, stream>>>(esrc, edst, als, ald, emax, E, Etot);
  gat_edge_sum<<<gEdge, B, 0, stream>>>(esrc, edst, als, ald, emax, denom, E, Etot);
  gat_edge_agg<<<aggBlocks, B, 0, stream>>>(esrc, edst, als, ald, emax, denom,
                                            h, agg, E, Etot, nWaves);
  gat_bias_elu<<<gFin1, B, 0, stream>>>(agg, b1, N);   // agg now holds ELU(h1)

  // ---- Layer 2 (mean over heads) ----
  gat_gemm128_wmma<<<gGemm, B, 0, stream>>>(agg, W2, h, N);
  gat_attn_logits<<<gNodeW, B, 0, stream>>>(h, as2, ad2, als, ald, N);
  gat_init<<<gInit, B, 0, stream>>>(agg, emax, denom, N);  // layer-1 acts consumed
  gat_edge_max<<<gEdge, B, 0, stream>>>(esrc, edst, als, ald, emax, E, Etot);
  gat_edge_sum<<<gEdge, B, 0, stream>>>(esrc, edst, als, ald, emax, denom, E, Etot);
  gat_edge_agg<<<aggBlocks, B, 0, stream>>>(esrc, edst, als, ald, emax, denom,
                                            h, agg, E, Etot, nWaves);
  gat_head_mean_bias<<<gFin2, B, 0, stream>>>(agg, b2, (float*)d_out, N);
}